// IGAE_encoder_53626961657926
// MI455X (gfx1250) — compile-verified
//
#include <hip/hip_runtime.h>
#include <math.h>

typedef float v2f __attribute__((ext_vector_type(2)));
typedef float v8f __attribute__((ext_vector_type(8)));

#define NNODES 12000
#define NEDGES 384000
#define DIN    512
#define DH1    128
#define DH2    256
#define DH3    20
#define DH3P   32      // padded leading dim for layer-3 support / W3

// ---------------------------------------------------------------- zero fill
__global__ void zero_kernel(float* __restrict__ p, int n) {
    int i = blockIdx.x * blockDim.x + threadIdx.x;
    if (i < n) p[i] = 0.0f;
}

// ---------------------------------------------------------------- pad W3 [256,20] -> [256,32] (zero-filled)
__global__ void pad_w3_kernel(const float* __restrict__ W3, float* __restrict__ W3p) {
    int i = blockIdx.x * blockDim.x + threadIdx.x;
    if (i < DH2 * DH3P) {
        int r = i >> 5, c = i & (DH3P - 1);
        W3p[i] = (c < DH3) ? W3[r * DH3 + c] : 0.0f;
    }
}

// ---------------------------------------------------------------- dense GEMM (tail-free)
// D = act(A @ B); A[M,K] (lda), B[K,Nout] (ldb), D[M,Nout] (ldd), all row-major.
// M, Nout multiples of 16; K multiple of 4. One wave per 16x16 tile, fp32 WMMA.
// Inner loop: 1x global_load_b64 (A pair) + 2x global_load_b32 (B) + 1x wmma.
__global__ void gemm_f32_wmma(const float* __restrict__ A, int lda,
                              const float* __restrict__ B, int ldb,
                              float* __restrict__ Dm, int ldd,
                              int M, int K, int Nout, int act) {
    const int lane    = threadIdx.x & 31;
    const int wave    = threadIdx.x >> 5;
    const int ntilesN = Nout >> 4;
    const int ntiles  = (M >> 4) * ntilesN;
    const int tile    = blockIdx.x * (blockDim.x >> 5) + wave;
    if (tile >= ntiles) return;                  // wave-uniform exit
    const int tm = tile / ntilesN;
    const int tn = tile - tm * ntilesN;
    const int m0 = tm << 4, n0 = tn << 4;

    const int rowA  = m0 + (lane & 15);          // A row for this lane
    const int colB  = n0 + (lane & 15);          // B/D column for this lane
    const int khalf = (lane >> 4) << 1;          // lanes 0-15 -> K+0,+1 ; 16-31 -> K+2,+3

    const float* Ap = A + (size_t)rowA * lda + khalf;   // step +4 per iter
    const float* Bp = B + (size_t)khalf * ldb + colB;   // step +4*ldb per iter

    v8f acc = {};
    for (int k0 = 0; k0 < K; k0 += 4) {
        v2f a = *(const v2f*)Ap;                 // 8B-aligned: lda, khalf even
        v2f b;
        b.x = Bp[0];
        b.y = Bp[ldb];
        acc = __builtin_amdgcn_wmma_f32_16x16x4_f32(false, a, false, b,
                                                    (short)0, acc, false, false);
        Ap += 4;
        Bp += (size_t)4 * ldb;
    }
    const int rbase = m0 + ((lane >> 4) << 3);
#pragma unroll
    for (int v = 0; v < 8; ++v) {
        float d = acc[v];
        if (act) d = tanhf(d);
        Dm[(size_t)(rbase + v) * ldd + colB] = d;
    }
}

// ---------------------------------------------------------------- SpMM (scatter)
// out[row[e], 0:D] += val[e] * dense[col[e], 0:D]   via f32 global atomics.
// dense has leading dim lddense (>= D); out is dense-packed with leading dim D.
__global__ void spmm_atomic(const int* __restrict__ row, const int* __restrict__ col,
                            const float* __restrict__ val, const float* __restrict__ dense,
                            int lddense, float* __restrict__ out, int D) {
    const int e = blockIdx.x * (blockDim.x >> 5) + (threadIdx.x >> 5);
    if (e >= NEDGES) return;
    const int   lane = threadIdx.x & 31;
    const int   r = row[e];
    const int   c = col[e];
    const float v = val[e];
    for (int f = lane; f < D; f += 32)
        atomicAdd(&out[(size_t)r * D + f], v * dense[(size_t)c * lddense + f]);
}

// ---------------------------------------------------------------- Gram + sigmoid
// Out = sigmoid(Z @ Z^T), Z[NNODES, 20] row-major. One wave per 16x16 tile,
// exactly 5 fp32 WMMAs; bandwidth-bound on the 576 MB store.
__global__ void gram_sigmoid_wmma(const float* __restrict__ Z, float* __restrict__ Out) {
    const int lane = threadIdx.x & 31;
    const int wave = threadIdx.x >> 5;
    const int TN   = NNODES >> 4;                // 750
    const int tile = blockIdx.x * (blockDim.x >> 5) + wave;
    if (tile >= TN * TN) return;                 // wave-uniform exit
    const int tm = tile / TN;
    const int tn = tile - tm * TN;

    const int rowA  = (tm << 4) + (lane & 15);   // A row   (Z row)
    const int rowB  = (tn << 4) + (lane & 15);   // B "col" (Z row, since B = Z^T)
    const int khalf = (lane >> 4) << 1;

    const float* Za = Z + (size_t)rowA * DH3 + khalf;
    const float* Zb = Z + (size_t)rowB * DH3 + khalf;

    v8f acc = {};
#pragma unroll
    for (int k0 = 0; k0 < DH3; k0 += 4) {
        v2f a = *(const v2f*)(Za + k0);          // 8B-aligned (20, khalf even)
        v2f b = *(const v2f*)(Zb + k0);
        acc = __builtin_amdgcn_wmma_f32_16x16x4_f32(false, a, false, b,
                                                    (short)0, acc, false, false);
    }
    const int colD  = (tn << 4) + (lane & 15);
    const int rbase = (tm << 4) + ((lane >> 4) << 3);
#pragma unroll
    for (int v = 0; v < 8; ++v) {
        const float d = acc[v];
        Out[(size_t)(rbase + v) * NNODES + colD] = 1.0f / (1.0f + __expf(-d));
    }
}

// ---------------------------------------------------------------- launch
extern "C" void kernel_launch(void* const* d_in, const int* in_sizes, int n_in,
                              void* d_out, int out_size, void* d_ws, size_t ws_size,
                              hipStream_t stream) {
    (void)in_sizes; (void)n_in; (void)out_size; (void)ws_size;

    const float* x       = (const float*)d_in[0];
    const int*   adj_row = (const int*)  d_in[1];
    const int*   adj_col = (const int*)  d_in[2];
    const float* adj_val = (const float*)d_in[3];
    const float* W1      = (const float*)d_in[4];
    const float* W2      = (const float*)d_in[5];
    const float* W3      = (const float*)d_in[6];

    float* out = (float*)d_out;
    // Output layout: concatenated flat in reference return order.
    float* z_igae  = out;                               // N*20
    float* adj_out = out + 240000;                      // N*N
    float* az1     = out + 144240000;                   // N*128
    float* az2     = out + 145776000;                   // N*256
    float* az3     = out + 148848000;                   // N*20
    float* z1      = out + 149088000;                   // N*128
    float* z2      = out + 150624000;                   // N*256

    float* ws  = (float*)d_ws;
    float* s1  = ws;                                    // N*128  support1
    float* s2  = s1 + (size_t)NNODES * DH1;             // N*256  support2
    float* s3p = s2 + (size_t)NNODES * DH2;             // N*32   support3 (padded ld)
    float* W3p = s3p + (size_t)NNODES * DH3P;           // 256*32 padded W3

    // Re-zero atomically-accumulated regions every call (deterministic replay).
    auto zero = [&](float* p, int n) {
        zero_kernel<<<(n + 255) / 256, 256, 0, stream>>>(p, n);
    };
    zero(z1,     NNODES * DH1);  zero(az1, NNODES * DH1);
    zero(z2,     NNODES * DH2);  zero(az2, NNODES * DH2);
    zero(z_igae, NNODES * DH3);  zero(az3, NNODES * DH3);

    pad_w3_kernel<<<(DH2 * DH3P + 255) / 256, 256, 0, stream>>>(W3, W3p);

    const int spmmBlocks = NEDGES / 8;  // 8 waves (edges) per 256-thread block

    // ---- layer 1: s1 = tanh(x@W1); z1 = A@s1; az1 = A@z1
    {
        int tiles = (NNODES / 16) * (DH1 / 16);
        gemm_f32_wmma<<<(tiles + 7) / 8, 256, 0, stream>>>(
            x, DIN, W1, DH1, s1, DH1, NNODES, DIN, DH1, 1);
        spmm_atomic<<<spmmBlocks, 256, 0, stream>>>(adj_row, adj_col, adj_val, s1, DH1, z1, DH1);
        spmm_atomic<<<spmmBlocks, 256, 0, stream>>>(adj_row, adj_col, adj_val, z1, DH1, az1, DH1);
    }
    // ---- layer 2: s2 = tanh(z1@W2); z2 = A@s2; az2 = A@z2
    {
        int tiles = (NNODES / 16) * (DH2 / 16);
        gemm_f32_wmma<<<(tiles + 7) / 8, 256, 0, stream>>>(
            z1, DH1, W2, DH2, s2, DH2, NNODES, DH1, DH2, 1);
        spmm_atomic<<<spmmBlocks, 256, 0, stream>>>(adj_row, adj_col, adj_val, s2, DH2, z2, DH2);
        spmm_atomic<<<spmmBlocks, 256, 0, stream>>>(adj_row, adj_col, adj_val, z2, DH2, az2, DH2);
    }
    // ---- layer 3 (no activation, padded to 32 cols -> tail-free):
    //      s3p = z2@W3p; z_igae = A@s3p[:, :20]; az3 = A@z_igae
    {
        int tiles = (NNODES / 16) * (DH3P / 16);        // 750 * 2
        gemm_f32_wmma<<<(tiles + 7) / 8, 256, 0, stream>>>(
            z2, DH2, W3p, DH3P, s3p, DH3P, NNODES, DH2, DH3P, 0);
        spmm_atomic<<<spmmBlocks, 256, 0, stream>>>(adj_row, adj_col, adj_val, s3p, DH3P, z_igae, DH3);
        spmm_atomic<<<spmmBlocks, 256, 0, stream>>>(adj_row, adj_col, adj_val, z_igae, DH3, az3, DH3);
    }
    // ---- z_igae_adj = sigmoid(z_igae @ z_igae^T)
    {
        int tiles = (NNODES / 16) * (NNODES / 16);      // 562,500 tiles
        gram_sigmoid_wmma<<<(tiles + 7) / 8, 256, 0, stream>>>(z_igae, adj_out);
    }
}